// SPA_37924561224321
// MI455X (gfx1250) — compile-verified
//
#include <hip/hip_runtime.h>
#include <hip/hip_bf16.h>
#include <math.h>

#define NNODES 50000
#define NEDGES 800000
#define CH 128
#define OUTCH 64
#define STAT_BLOCKS 512

typedef __attribute__((ext_vector_type(2))) float v2f;
typedef __attribute__((ext_vector_type(8))) float v8f;

// ---------- helpers ----------
__device__ __forceinline__ float lrelu01(float a) {
  return (a > 0.0f ? a : 0.2f * a) * 0.1f;
}
// monotone float -> uint encoding so unsigned atomicMax == float max
__device__ __forceinline__ unsigned fenc(float f) {
  unsigned u = __float_as_uint(f);
  return (u & 0x80000000u) ? ~u : (u | 0x80000000u);
}
__device__ __forceinline__ float fdec(unsigned k) {
  unsigned u = (k & 0x80000000u) ? (k ^ 0x80000000u) : ~k;
  return __uint_as_float(u);
}

// ---------- WMMA f32 GEMM: Y[N,M] = X[N,K] * W[M,K]^T + bias ----------
// one wave per 16x16 output tile; V_WMMA_F32_16X16X4_F32, K-loop step 4
__global__ void gemm_wmma_f32(const float* __restrict__ X,
                              const float* __restrict__ W,
                              const float* __restrict__ bias,
                              float* __restrict__ Y,
                              int Nrows, int K, int M) {
  int wave = (int)((blockIdx.x * blockDim.x + threadIdx.x) >> 5);
  int lane = threadIdx.x & 31;
  int tilesN = M >> 4;
  int totalTiles = (Nrows >> 4) * tilesN;
  if (wave >= totalTiles) return;           // whole-wave exit: EXEC stays all-1s
  int tm = wave / tilesN;
  int tn = wave % tilesN;
  int hl = lane >> 4;                       // half-wave select
  int l  = lane & 15;
  const float* xrow = X + (size_t)(tm * 16 + l) * K;
  const float* wrow = W + (size_t)(tn * 16 + l) * K;
  v8f c = {};
  for (int k = 0; k < K; k += 4) {
    int kk = k + hl * 2;                    // ISA layout: VGPR j holds K = 2*half + j
    v2f a, b;
    a.x = xrow[kk];     a.y = xrow[kk + 1];
    b.x = wrow[kk];     b.y = wrow[kk + 1];
    c = __builtin_amdgcn_wmma_f32_16x16x4_f32(
        /*neg_a=*/false, a, /*neg_b=*/false, b,
        /*c_mod=*/(short)0, c, /*reuse_a=*/false, /*reuse_b=*/false);
  }
  int col = tn * 16 + l;
  float bv = bias[col];
  int rowbase = tm * 16 + hl * 8;           // C/D layout: M = r + 8*half, N = lane&15
  for (int r = 0; r < 8; ++r)
    Y[(size_t)(rowbase + r) * M + col] = c[r] + bv;
}

// ---------- edge moment statistics: partial sums (deterministic) ----------
__global__ void edge_stats(const float* __restrict__ h,
                           const int* __restrict__ src,
                           const int* __restrict__ dst,
                           int E, double* __restrict__ partials) {
  int c = threadIdx.x;                      // 0..127 (channel)
  int b = blockIdx.x;
  int per = (E + gridDim.x - 1) / gridDim.x;
  int e0 = b * per;
  int e1 = e0 + per; if (e1 > E) e1 = E;
  double sum = 0.0, sq = 0.0;
  for (int e = e0; e < e1; ++e) {
    int s = src[e], d = dst[e];
    float diff = h[(size_t)s * CH + c] - h[(size_t)d * CH + c];
    float v = diff * diff;
    sum += (double)v;
    sq  += (double)v * (double)v;
  }
  partials[((size_t)b * CH + c) * 2 + 0] = sum;
  partials[((size_t)b * CH + c) * 2 + 1] = sq;
}

// ---------- finalize stats + tiny attention solve -> att_l/att_r [CH] ----------
__global__ void stats_att(const double* __restrict__ partials, int nblk, int E,
                          const float* __restrict__ src_w, const float* __restrict__ src_b,
                          const float* __restrict__ dst_w, const float* __restrict__ dst_b,
                          const float* __restrict__ tq,
                          float* __restrict__ att_l, float* __restrict__ att_r) {
  int c = threadIdx.x;                      // 0..127
  double sum = 0.0, sq = 0.0;
  for (int b = 0; b < nblk; ++b) {
    sum += partials[((size_t)b * CH + c) * 2 + 0];
    sq  += partials[((size_t)b * CH + c) * 2 + 1];
  }
  double m1d  = sum / (double)E;
  double var  = (sq - sum * sum / (double)E) / (double)(E - 1);
  if (var < 0.0) var = 0.0;
  double stdd = sqrt(var);
  double m2d  = stdd + 1e-5;
  float S[4];
  S[0] = (float)m1d;
  S[1] = (float)stdd;
  S[2] = (float)((m1d * m1d * m1d) / (m2d * m2d * m2d));
  S[3] = (float)((m1d * m1d) * (m1d * m1d) / ((m2d * m2d) * (m2d * m2d)));
  __shared__ float Ssh[4][CH];
  __shared__ float norms[4];
  for (int s = 0; s < 4; ++s) {
    float v = S[s];
    if (v != v) v = 0.0f;                   // NaN -> 0 (before tanh, as reference)
    v = tanhf(v);
    S[s] = v;
    Ssh[s][c] = v;
  }
  __syncthreads();
  if (c < 4) {
    float acc = 0.0f;
    for (int j = 0; j < CH; ++j) acc += Ssh[c][j] * Ssh[c][j];
    norms[c] = fmaxf(sqrtf(acc), 1e-12f);
  }
  __syncthreads();
  // att = (S @ w^T + b) @ tq  ==  const + sum_s S[s]*v[s]
  float vl[4] = {0,0,0,0}, vr[4] = {0,0,0,0};
  float Cl = 0.0f, Cr = 0.0f;
  for (int j = 0; j < 16; ++j) {
    float t = tq[j];
    Cl += t * src_b[j];
    Cr += t * dst_b[j];
    for (int s = 0; s < 4; ++s) {
      vl[s] += t * src_w[j * 4 + s];
      vr[s] += t * dst_w[j * 4 + s];
    }
  }
  float al = Cl, ar = Cr;
  for (int s = 0; s < 4; ++s) {
    float Sn = S[s] / norms[s];
    al += Sn * vl[s];
    ar += Sn * vr[s];
  }
  att_l[c] = al;
  att_r[c] = ar;
}

// ---------- per-node scores: a_src = h@att_l, a_dst = h@att_r; init max w/ self loop ----------
__global__ void node_scores(const float* __restrict__ h,
                            const float* __restrict__ att_l,
                            const float* __restrict__ att_r,
                            float* __restrict__ a_srcv, float* __restrict__ a_dstv,
                            unsigned* __restrict__ m_enc, int Nn) {
  int wave = (int)((blockIdx.x * blockDim.x + threadIdx.x) >> 5);
  int lane = threadIdx.x & 31;
  if (wave >= Nn) return;
  const float4* row = (const float4*)(h + (size_t)wave * CH);
  const float4* al  = (const float4*)att_l;
  const float4* ar  = (const float4*)att_r;
  float4 x4 = row[lane];                    // 128 floats = 32 float4, one per lane
  float4 a4 = al[lane];
  float4 r4 = ar[lane];
  float sl = x4.x*a4.x + x4.y*a4.y + x4.z*a4.z + x4.w*a4.w;
  float sr = x4.x*r4.x + x4.y*r4.y + x4.z*r4.z + x4.w*r4.w;
  for (int off = 16; off > 0; off >>= 1) {
    sl += __shfl_down(sl, off, 32);
    sr += __shfl_down(sr, off, 32);
  }
  if (lane == 0) {
    a_srcv[wave] = sl;
    a_dstv[wave] = sr;
    m_enc[wave]  = fenc(lrelu01(sl + sr)); // self-loop seeds the segment max
  }
}

// ---------- segment max over incoming edges ----------
__global__ void edge_max(const int* __restrict__ src, const int* __restrict__ dst,
                         const float* __restrict__ a_srcv, const float* __restrict__ a_dstv,
                         unsigned* __restrict__ m_enc, int E) {
  int e = blockIdx.x * blockDim.x + threadIdx.x;
  if (e >= E) return;
  int s = src[e], d = dst[e];
  float a = lrelu01(a_srcv[s] + a_dstv[d]);
  atomicMax(&m_enc[d], fenc(a));
}

// ---------- decode max, seed sum with self-loop exp ----------
__global__ void finalize_max_init_sum(const float* __restrict__ a_srcv,
                                      const float* __restrict__ a_dstv,
                                      const unsigned* __restrict__ m_enc,
                                      float* __restrict__ m_f, float* __restrict__ s_sum,
                                      int Nn) {
  int n = blockIdx.x * blockDim.x + threadIdx.x;
  if (n >= Nn) return;
  float m = fdec(m_enc[n]);
  m_f[n] = m;
  float a = lrelu01(a_srcv[n] + a_dstv[n]);
  s_sum[n] = expf(a - m);
}

// ---------- segment sum of exp ----------
__global__ void edge_exp_sum(const int* __restrict__ src, const int* __restrict__ dst,
                             const float* __restrict__ a_srcv, const float* __restrict__ a_dstv,
                             const float* __restrict__ m_f, float* __restrict__ s_sum, int E) {
  int e = blockIdx.x * blockDim.x + threadIdx.x;
  if (e >= E) return;
  int s = src[e], d = dst[e];
  float a = lrelu01(a_srcv[s] + a_dstv[d]);
  atomicAdd(&s_sum[d], expf(a - m_f[d]));
}

// ---------- out init: self-loop contribution + conv bias ----------
__global__ void agg_init(const float* __restrict__ h,
                         const float* __restrict__ a_srcv, const float* __restrict__ a_dstv,
                         const float* __restrict__ m_f, const float* __restrict__ s_sum,
                         const float* __restrict__ bias, float* __restrict__ out, int Nn) {
  int idx = blockIdx.x * blockDim.x + threadIdx.x;
  if (idx >= Nn * CH) return;
  int n = idx >> 7;
  int c = idx & (CH - 1);
  float a = lrelu01(a_srcv[n] + a_dstv[n]);
  float alpha = expf(a - m_f[n]) / (s_sum[n] + 1e-16f);
  out[idx] = alpha * h[idx] + bias[c];
}

// ---------- scatter weighted features along edges (wave per edge) ----------
__global__ void agg_edges(const float* __restrict__ h,
                          const int* __restrict__ src, const int* __restrict__ dst,
                          const float* __restrict__ a_srcv, const float* __restrict__ a_dstv,
                          const float* __restrict__ m_f, const float* __restrict__ s_sum,
                          float* __restrict__ out, int E) {
  int wave = (int)((blockIdx.x * blockDim.x + threadIdx.x) >> 5);
  int lane = threadIdx.x & 31;
  if (wave >= E) return;
  int s = src[wave], d = dst[wave];
  float a = lrelu01(a_srcv[s] + a_dstv[d]);
  float alpha = expf(a - m_f[d]) / (s_sum[d] + 1e-16f);
  const float4* hs = (const float4*)(h + (size_t)s * CH);
  float4 v = hs[lane];
  float* o = out + (size_t)d * CH + lane * 4;
  atomicAdd(o + 0, alpha * v.x);
  atomicAdd(o + 1, alpha * v.y);
  atomicAdd(o + 2, alpha * v.z);
  atomicAdd(o + 3, alpha * v.w);
}

__global__ void relu_kernel(float* __restrict__ x, int n) {
  int i = blockIdx.x * blockDim.x + threadIdx.x;
  if (i < n) x[i] = fmaxf(x[i], 0.0f);
}

// ---------- host-side conv sequence ----------
static void run_conv(const float* h_in, float* h_out,
                     const int* src, const int* dst,
                     const float* src_w, const float* src_b,
                     const float* dst_w, const float* dst_b,
                     const float* tq, const float* bias,
                     double* partials, float* att_l, float* att_r,
                     float* a_srcv, float* a_dstv, unsigned* m_enc,
                     float* m_f, float* s_sum, hipStream_t stream) {
  edge_stats<<<STAT_BLOCKS, CH, 0, stream>>>(h_in, src, dst, NEDGES, partials);
  stats_att<<<1, CH, 0, stream>>>(partials, STAT_BLOCKS, NEDGES,
                                  src_w, src_b, dst_w, dst_b, tq, att_l, att_r);
  node_scores<<<(NNODES * 32 + 255) / 256, 256, 0, stream>>>(h_in, att_l, att_r,
                                                             a_srcv, a_dstv, m_enc, NNODES);
  edge_max<<<(NEDGES + 255) / 256, 256, 0, stream>>>(src, dst, a_srcv, a_dstv, m_enc, NEDGES);
  finalize_max_init_sum<<<(NNODES + 255) / 256, 256, 0, stream>>>(a_srcv, a_dstv, m_enc,
                                                                  m_f, s_sum, NNODES);
  edge_exp_sum<<<(NEDGES + 255) / 256, 256, 0, stream>>>(src, dst, a_srcv, a_dstv,
                                                         m_f, s_sum, NEDGES);
  agg_init<<<(NNODES * CH + 255) / 256, 256, 0, stream>>>(h_in, a_srcv, a_dstv,
                                                          m_f, s_sum, bias, h_out, NNODES);
  agg_edges<<<(NEDGES * 32 + 255) / 256, 256, 0, stream>>>(h_in, src, dst, a_srcv, a_dstv,
                                                           m_f, s_sum, h_out, NEDGES);
}

extern "C" void kernel_launch(void* const* d_in, const int* in_sizes, int n_in,
                              void* d_out, int out_size, void* d_ws, size_t ws_size,
                              hipStream_t stream) {
  const float* x        = (const float*)d_in[0];
  const int*   ei       = (const int*)d_in[1];
  const float* W0       = (const float*)d_in[2];
  const float* b0       = (const float*)d_in[3];
  const float* W2       = (const float*)d_in[4];
  const float* b2       = (const float*)d_in[5];
  const float* g0_src_w = (const float*)d_in[6];
  const float* g0_src_b = (const float*)d_in[7];
  const float* g0_dst_w = (const float*)d_in[8];
  const float* g0_dst_b = (const float*)d_in[9];
  const float* g0_tq    = (const float*)d_in[10];
  const float* g0_bias  = (const float*)d_in[11];
  const float* g2_src_w = (const float*)d_in[12];
  const float* g2_src_b = (const float*)d_in[13];
  const float* g2_dst_w = (const float*)d_in[14];
  const float* g2_dst_b = (const float*)d_in[15];
  const float* g2_tq    = (const float*)d_in[16];
  const float* g2_bias  = (const float*)d_in[17];

  const int* src = ei;              // edge_index[0]
  const int* dst = ei + NEDGES;     // edge_index[1]

  // workspace layout (all offsets 16B aligned)
  char* ws = (char*)d_ws;
  size_t off = 0;
  float* hA = (float*)(ws + off);                 off += (size_t)NNODES * CH * 4;   // 25.6MB
  float* hB = (float*)(ws + off);                 off += (size_t)NNODES * CH * 4;   // 25.6MB
  double* partials = (double*)(ws + off);         off += (size_t)STAT_BLOCKS * CH * 2 * 8;
  float* att_l = (float*)(ws + off);              off += 512;
  float* att_r = (float*)(ws + off);              off += 512;
  float* a_srcv = (float*)(ws + off);             off += (size_t)NNODES * 4;
  float* a_dstv = (float*)(ws + off);             off += (size_t)NNODES * 4;
  unsigned* m_enc = (unsigned*)(ws + off);        off += (size_t)NNODES * 4;
  float* m_f = (float*)(ws + off);                off += (size_t)NNODES * 4;
  float* s_sum = (float*)(ws + off);              off += (size_t)NNODES * 4;
  (void)off; (void)ws_size; (void)in_sizes; (void)n_in; (void)out_size;

  // 1) h = x @ W0^T + b0   (WMMA f32)
  {
    int tiles = (NNODES / 16) * (CH / 16);        // 25000 tiles, 8 waves/block
    gemm_wmma_f32<<<(tiles + 7) / 8, 256, 0, stream>>>(x, W0, b0, hA, NNODES, CH, CH);
  }
  // 2) conv g0 -> hB, then ReLU
  run_conv(hA, hB, src, dst, g0_src_w, g0_src_b, g0_dst_w, g0_dst_b, g0_tq, g0_bias,
           partials, att_l, att_r, a_srcv, a_dstv, m_enc, m_f, s_sum, stream);
  relu_kernel<<<(NNODES * CH + 255) / 256, 256, 0, stream>>>(hB, NNODES * CH);
  // 3) conv g2 -> hA
  run_conv(hB, hA, src, dst, g2_src_w, g2_src_b, g2_dst_w, g2_dst_b, g2_tq, g2_bias,
           partials, att_l, att_r, a_srcv, a_dstv, m_enc, m_f, s_sum, stream);
  // 4) out = hA @ W2^T + b2  (WMMA f32, M=64)
  {
    int tiles = (NNODES / 16) * (OUTCH / 16);     // 12500 tiles
    gemm_wmma_f32<<<(tiles + 7) / 8, 256, 0, stream>>>(hA, W2, b2, (float*)d_out,
                                                       NNODES, CH, OUTCH);
  }
}